// NonLinearity_27230092657495
// MI455X (gfx1250) — compile-verified
//
#include <hip/hip_runtime.h>
#include <stdint.h>
#include <stddef.h>

#define NU    256   // num_units
#define NIN   128   // input size
#define BT    8     // batch rows per block
#define NFOLD 12

static constexpr float kEPS = 1e-8f;

// ---------------- CDNA5 fast tanh (v_tanh_f32, new on gfx1250) -------------
__device__ __forceinline__ float fast_tanh(float x) {
#if __has_builtin(__builtin_amdgcn_tanhf)
  return __builtin_amdgcn_tanhf(x);
#elif __has_builtin(__builtin_amdgcn_tanh_f32)
  return __builtin_amdgcn_tanh_f32(x);
#else
  // tanh(x) = 2*sigmoid(2x) - 1  (2 trans ops fallback)
  return __builtin_fmaf(2.0f, __frcp_rn(1.0f + __expf(-2.0f * x)), -1.0f);
#endif
}

// ---------------- CDNA5 async global->LDS copy (ASYNCcnt path) -------------
typedef __attribute__((address_space(1))) int as1_int;
typedef __attribute__((address_space(3))) int as3_int;

__device__ __forceinline__ void async_copy_b32(const float* g, float* l) {
#if __has_builtin(__builtin_amdgcn_global_load_async_to_lds_b32)
  __builtin_amdgcn_global_load_async_to_lds_b32(
      (as1_int*)(g), (as3_int*)(l), 0, 0);
#else
  *l = *g;
#endif
}

__device__ __forceinline__ void wait_async0() {
#if __has_builtin(__builtin_amdgcn_s_wait_asynccnt)
  __builtin_amdgcn_s_wait_asynccnt(0);
#elif defined(__AMDGCN__)
  asm volatile("s_wait_asynccnt 0" ::: "memory");
#endif
}

// ---------------- fused LTC fold body --------------------------------------
template <bool PACKED>
__device__ __forceinline__ void ltc_fold_body(
    const float* __restrict__ inputs, const float* __restrict__ state,
    const float4* __restrict__ ps,    // packed sensory [NIN*NU] (PACKED)
    const float4* __restrict__ pr,    // packed recurrent [NU*NU] (PACKED)
    const float* __restrict__ smu, const float* __restrict__ ssg,
    const float* __restrict__ sw,  const float* __restrict__ ser,
    const float* __restrict__ rmu, const float* __restrict__ rsg,
    const float* __restrict__ rw,  const float* __restrict__ rer,
    const float* __restrict__ vleak, const float* __restrict__ gleak,
    const float* __restrict__ cm_t,  float* __restrict__ out) {
  __shared__ float v_s[BT * NU];    // 8 KB: state tile, broadcast-read
  __shared__ float in_s[BT * NIN];  // 4 KB: input tile

  const int tx = threadIdx.x;       // unit index u
  const int b0 = blockIdx.x * BT;   // first batch row of this block

  // Stage the input tile via CDNA5 async global->LDS loads.
#pragma unroll
  for (int k = 0; k < (BT * NIN) / 256; ++k) {
    int idx = tx + 256 * k;
    int b = idx >> 7, i = idx & (NIN - 1);
    async_copy_b32(&inputs[(size_t)(b0 + b) * NIN + i], &in_s[idx]);
  }
  // Stage the state tile (normal coalesced loads).
#pragma unroll
  for (int b = 0; b < BT; ++b)
    v_s[b * NU + tx] = state[(size_t)(b0 + b) * NU + tx];

  const float cm   = cm_t[tx];
  const float gl   = gleak[tx];
  const float glvl = gl * vleak[tx];
  const float cge  = cm + gl + kEPS;   // constant part of denominator

  wait_async0();
  __syncthreads();

  // ---- sensory sums (done once) ----
  float den_s[BT], num_s[BT];
#pragma unroll
  for (int b = 0; b < BT; ++b) { den_s[b] = 0.0f; num_s[b] = 0.0f; }

#pragma unroll 4
  for (int i = 0; i < NIN; ++i) {
    float sgh, sm, wh, er;
    if (PACKED) {
      float4 p = ps[i * NU + tx];
      sgh = p.x; sm = p.y; wh = p.z; er = p.w;
    } else {
      float m = smu[i * NU + tx];
      sgh = 0.5f * ssg[i * NU + tx];
      sm  = sgh * m;
      wh  = 0.5f * sw[i * NU + tx];
      er  = ser[i * NU + tx];
    }
#pragma unroll
    for (int b = 0; b < BT; ++b) {
      // W*sigmoid(sigma*(x-mu)) = wh*(1 + tanh(sgh*x - sm))
      float x  = __builtin_fmaf(sgh, in_s[b * NIN + i], -sm);
      float g  = __builtin_fmaf(wh, fast_tanh(x), wh);
      den_s[b] += g;
      num_s[b]  = __builtin_fmaf(g, er, num_s[b]);
    }
  }

  // ---- 12 fold iterations ----
  for (int fold = 0; fold < NFOLD; ++fold) {
    float wr[BT], wn[BT];
#pragma unroll
    for (int b = 0; b < BT; ++b) {
      wr[b] = den_s[b] + cge;    // cm + gleak + eps + w_den_sensory
      wn[b] = num_s[b] + glvl;   // gleak*vleak + w_num_sensory
    }
#pragma unroll 4
    for (int j = 0; j < NU; ++j) {
      float sgh, sm, wh, er;
      if (PACKED) {
        float4 p = pr[j * NU + tx];
        sgh = p.x; sm = p.y; wh = p.z; er = p.w;
      } else {
        float m = rmu[j * NU + tx];
        sgh = 0.5f * rsg[j * NU + tx];
        sm  = sgh * m;
        wh  = 0.5f * rw[j * NU + tx];
        er  = rer[j * NU + tx];
      }
#pragma unroll
      for (int b = 0; b < BT; ++b) {
        float x = __builtin_fmaf(sgh, v_s[b * NU + j], -sm);  // LDS broadcast
        float g = __builtin_fmaf(wh, fast_tanh(x), wh);
        wr[b] += g;
        wn[b]  = __builtin_fmaf(g, er, wn[b]);
      }
    }
    float vn[BT];
#pragma unroll
    for (int b = 0; b < BT; ++b)
      vn[b] = __builtin_fmaf(cm, v_s[b * NU + tx], wn[b]) / wr[b];
    __syncthreads();  // all reads of old v done
#pragma unroll
    for (int b = 0; b < BT; ++b)
      v_s[b * NU + tx] = vn[b];
    __syncthreads();  // all writes of new v visible
  }

#pragma unroll
  for (int b = 0; b < BT; ++b)
    out[(size_t)(b0 + b) * NU + tx] = v_s[b * NU + tx];
}

// ---- hot kernel defined FIRST so the disasm snippet shows its code --------
__global__ __launch_bounds__(256) void ltc_fold_packed(
    const float* __restrict__ inputs, const float* __restrict__ state,
    const float4* __restrict__ ps, const float4* __restrict__ pr,
    const float* __restrict__ vleak, const float* __restrict__ gleak,
    const float* __restrict__ cm_t, float* __restrict__ out) {
  ltc_fold_body<true>(inputs, state, ps, pr,
                      nullptr, nullptr, nullptr, nullptr,
                      nullptr, nullptr, nullptr, nullptr,
                      vleak, gleak, cm_t, out);
}

// ---------------- parameter pre-pack: {0.5*sigma, 0.5*sigma*mu, 0.5*W, erev}
__global__ __launch_bounds__(256) void pack_params_kernel(
    const float* __restrict__ mu, const float* __restrict__ sg,
    const float* __restrict__ w,  const float* __restrict__ er,
    float4* __restrict__ out, int n) {
  int i = blockIdx.x * 256 + threadIdx.x;
  if (i < n) {
    float sgh = 0.5f * sg[i];
    out[i] = make_float4(sgh, sgh * mu[i], 0.5f * w[i], er[i]);
  }
}

// ---- fallback (workspace too small): reads raw param arrays ---------------
__global__ __launch_bounds__(256) void ltc_fold_raw(
    const float* __restrict__ inputs, const float* __restrict__ state,
    const float* __restrict__ smu, const float* __restrict__ ssg,
    const float* __restrict__ sw,  const float* __restrict__ ser,
    const float* __restrict__ rmu, const float* __restrict__ rsg,
    const float* __restrict__ rw,  const float* __restrict__ rer,
    const float* __restrict__ vleak, const float* __restrict__ gleak,
    const float* __restrict__ cm_t, float* __restrict__ out) {
  ltc_fold_body<false>(inputs, state, nullptr, nullptr,
                       smu, ssg, sw, ser, rmu, rsg, rw, rer,
                       vleak, gleak, cm_t, out);
}

// ---------------- launch ----------------------------------------------------
extern "C" void kernel_launch(void* const* d_in, const int* in_sizes, int n_in,
                              void* d_out, int out_size, void* d_ws, size_t ws_size,
                              hipStream_t stream) {
  const float* inputs = (const float*)d_in[0];
  const float* state  = (const float*)d_in[1];
  const float* smu    = (const float*)d_in[2];
  const float* ssg    = (const float*)d_in[3];
  const float* sw     = (const float*)d_in[4];
  const float* ser    = (const float*)d_in[5];
  const float* rmu    = (const float*)d_in[6];
  const float* rsg    = (const float*)d_in[7];
  const float* rw     = (const float*)d_in[8];
  const float* rer    = (const float*)d_in[9];
  const float* vleak  = (const float*)d_in[10];
  const float* gleak  = (const float*)d_in[11];
  const float* cm_t   = (const float*)d_in[12];
  float* out = (float*)d_out;

  const int B = in_sizes[0] / NIN;  // 2048
  const int nRec = NU * NU;         // 65536
  const int nSen = NIN * NU;        // 32768
  const size_t packed_bytes = (size_t)(nRec + nSen) * sizeof(float4);  // 1.5 MB

  dim3 block(256);
  dim3 grid(B / BT);

  if (ws_size >= packed_bytes) {
    float4* pr = (float4*)d_ws;
    float4* ps = pr + nRec;
    pack_params_kernel<<<dim3((nRec + 255) / 256), block, 0, stream>>>(
        rmu, rsg, rw, rer, pr, nRec);
    pack_params_kernel<<<dim3((nSen + 255) / 256), block, 0, stream>>>(
        smu, ssg, sw, ser, ps, nSen);
    ltc_fold_packed<<<grid, block, 0, stream>>>(
        inputs, state, ps, pr, vleak, gleak, cm_t, out);
  } else {
    ltc_fold_raw<<<grid, block, 0, stream>>>(
        inputs, state, smu, ssg, sw, ser, rmu, rsg, rw, rer,
        vleak, gleak, cm_t, out);
  }
}